// GCNLayer_54073638257182
// MI455X (gfx1250) — compile-verified
//
#include <hip/hip_runtime.h>

#define NN 12288
#define DD 64
constexpr float BN_EPS = 1e-5f;
constexpr float SLOPE  = 0.01f;

typedef __attribute__((ext_vector_type(2))) float v2f;
typedef __attribute__((ext_vector_type(8))) float v8f;

// ---------------- Kernel 1: d[i] = rsqrt(rowsum(A) + 1) ----------------
__global__ __launch_bounds__(256) void k_rowsum(const float* __restrict__ A,
                                                float* __restrict__ dvec) {
    const int row = blockIdx.x;
    const float4* Ar = (const float4*)(A + (size_t)row * NN);
    float s = 0.f;
    for (int i = threadIdx.x; i < NN / 4; i += 256) {
        float4 v = Ar[i];
        s += v.x + v.y + v.z + v.w;
    }
    __shared__ float red[256];
    red[threadIdx.x] = s;
    __syncthreads();
    for (int off = 128; off > 0; off >>= 1) {
        if (threadIdx.x < off) red[threadIdx.x] += red[threadIdx.x + off];
        __syncthreads();
    }
    if (threadIdx.x == 0) dvec[row] = rsqrtf(red[0] + 1.0f);
}

// ------- Kernel 2: X = H@W + b ; Yt[j][i] = d[i]*X[i][j] (transposed) -------
__global__ __launch_bounds__(256) void k_linear(const float* __restrict__ H,
                                                const float* __restrict__ W,
                                                const float* __restrict__ b,
                                                const float* __restrict__ dvec,
                                                float* __restrict__ X,
                                                float* __restrict__ Yt) {
    __shared__ float Ws[64 * 64];
    __shared__ float Hs[4 * 64];
    const int t = threadIdx.x;
    const int i0 = blockIdx.x * 4;
#pragma unroll
    for (int u = 0; u < 4; ++u) {
        int idx = t + u * 256;                      // 1024 float4 = 4096 floats
        ((float4*)Ws)[idx] = ((const float4*)W)[idx];
    }
    if (t < 64) ((float4*)Hs)[t] = ((const float4*)(H + (size_t)i0 * DD))[t];
    __syncthreads();
    const int il = t >> 6;        // 0..3  (row within block)
    const int j  = t & 63;        // feature
    const int i  = i0 + il;
    float s = 0.f;
#pragma unroll
    for (int k = 0; k < 64; ++k) s += Hs[il * 64 + k] * Ws[k * 64 + j];
    const float xv = s + b[j];
    X[(size_t)i * DD + j] = xv;
    Yt[(size_t)j * NN + i] = dvec[i] * xv;
}

// ------- Kernel 3: Z = A @ Y via WMMA, async double-buffered staging -------
//         out_pre = d*Z + d^2*X
__global__ __launch_bounds__(256) void k_gemm(const float* __restrict__ A,
                                              const float* __restrict__ Yt,
                                              const float* __restrict__ X,
                                              const float* __restrict__ dvec,
                                              float* __restrict__ out_pre) {
    __shared__ float As[2][32 * 64];   // double-buffered A chunk (2 x 8 KB)
    __shared__ float Bs[2][64 * 64];   // double-buffered Yt chunk (2 x 16 KB)
    const int t     = threadIdx.x;
    const int lane  = t & 31;
    const int w     = t >> 5;                       // wave 0..7
    const int mbase = blockIdx.x * 32;
    const int arow  = (w >> 2) * 16 + (lane & 15);  // A row within As
    const int bcol  = (w & 3) * 16 + (lane & 15);   // output column (= Bs row)
    const int kbase = (lane >> 4) * 2;              // K sub-offset per ISA layout
    v8f acc = {};

    // Issue one chunk of async global->LDS copies (6 x b128 per thread/wave).
    auto stage = [&](int buf, int k0) {
#pragma unroll
        for (int u = 0; u < 2; ++u) {               // A: 32x64 = 512 float4
            int idx = t + u * 256;
            int r = idx >> 4, sg = idx & 15;
            unsigned la = (unsigned)(size_t)&As[buf][r * 64 + sg * 4];
            const float* ga = A + (size_t)(mbase + r) * NN + k0 + sg * 4;
            asm volatile("global_load_async_to_lds_b128 %0, %1, off"
                         :: "v"(la), "v"(ga) : "memory");
        }
#pragma unroll
        for (int u = 0; u < 4; ++u) {               // B: 64x64 = 1024 float4
            int idx = t + u * 256;
            int r = idx >> 4, sg = idx & 15;
            unsigned lb = (unsigned)(size_t)&Bs[buf][r * 64 + sg * 4];
            const float* gb = Yt + (size_t)r * NN + k0 + sg * 4;
            asm volatile("global_load_async_to_lds_b128 %0, %1, off"
                         :: "v"(lb), "v"(gb) : "memory");
        }
    };

    stage(0, 0);                                    // prologue: fill buffer 0
    for (int k0 = 0; k0 < NN; k0 += 64) {
        const int cur = (k0 >> 6) & 1;
        __syncthreads();   // all waves done reading buf cur^1 -> safe to overwrite
        if (k0 + 64 < NN) {
            stage(cur ^ 1, k0 + 64);                // prefetch next chunk
            // async loads complete in order: <=6 outstanding => current landed
            asm volatile("s_wait_asynccnt 0x6" ::: "memory");
        } else {
            asm volatile("s_wait_asynccnt 0x0" ::: "memory");
        }
        __syncthreads();   // every wave's async data for buf cur is in LDS
#pragma unroll
        for (int kk = 0; kk < 64; kk += 4) {
            v2f av = *(const v2f*)(&As[cur][arow * 64 + kk + kbase]);
            v2f bv = *(const v2f*)(&Bs[cur][bcol * 64 + kk + kbase]);
            acc = __builtin_amdgcn_wmma_f32_16x16x4_f32(
                false, av, false, bv, (short)0, acc, false, false);
        }
    }

    // Epilogue: C/D layout -> VGPR r: row = m0 + (lane>>4)*8 + r, col = lane&15
    const int col = bcol;
#pragma unroll
    for (int r = 0; r < 8; ++r) {
        int row = mbase + (w >> 2) * 16 + (lane >> 4) * 8 + r;
        float dr = dvec[row];
        float xv = X[(size_t)row * DD + col];
        out_pre[(size_t)row * DD + col] = dr * acc[r] + dr * dr * xv;
    }
}

// ------- Kernel 4: per-feature BN stats -> scale/shift -------
__global__ __launch_bounds__(256) void k_stats(const float* __restrict__ out_pre,
                                               const float* __restrict__ gamma,
                                               const float* __restrict__ beta,
                                               float* __restrict__ scale,
                                               float* __restrict__ shift) {
    const int col = blockIdx.x;
    float s = 0.f, s2 = 0.f;
    for (int i = threadIdx.x; i < NN; i += 256) {
        float v = out_pre[(size_t)i * DD + col];
        s += v; s2 += v * v;
    }
    __shared__ float r1[256], r2[256];
    r1[threadIdx.x] = s; r2[threadIdx.x] = s2;
    __syncthreads();
    for (int off = 128; off > 0; off >>= 1) {
        if (threadIdx.x < off) {
            r1[threadIdx.x] += r1[threadIdx.x + off];
            r2[threadIdx.x] += r2[threadIdx.x + off];
        }
        __syncthreads();
    }
    if (threadIdx.x == 0) {
        float mean = r1[0] * (1.0f / NN);
        float var  = r2[0] * (1.0f / NN) - mean * mean;   // biased, like torch BN
        float sc   = gamma[col] * rsqrtf(var + BN_EPS);
        scale[col] = sc;
        shift[col] = beta[col] - mean * sc;
    }
}

// ------- Kernel 5: BN affine + LeakyReLU, float4 -------
__global__ __launch_bounds__(256) void k_bn(const float* __restrict__ out_pre,
                                            const float* __restrict__ scale,
                                            const float* __restrict__ shift,
                                            float* __restrict__ out) {
    const int idx4 = blockIdx.x * 256 + threadIdx.x;   // 196608 float4 total
    float4 v = ((const float4*)out_pre)[idx4];
    const int j0 = (idx4 & 15) * 4;                    // 64 features / 4 per float4
    float a0 = v.x * scale[j0 + 0] + shift[j0 + 0];
    float a1 = v.y * scale[j0 + 1] + shift[j0 + 1];
    float a2 = v.z * scale[j0 + 2] + shift[j0 + 2];
    float a3 = v.w * scale[j0 + 3] + shift[j0 + 3];
    float4 r;
    r.x = a0 > 0.f ? a0 : SLOPE * a0;
    r.y = a1 > 0.f ? a1 : SLOPE * a1;
    r.z = a2 > 0.f ? a2 : SLOPE * a2;
    r.w = a3 > 0.f ? a3 : SLOPE * a3;
    ((float4*)out)[idx4] = r;
}

extern "C" void kernel_launch(void* const* d_in, const int* in_sizes, int n_in,
                              void* d_out, int out_size, void* d_ws, size_t ws_size,
                              hipStream_t stream) {
    const float* H     = (const float*)d_in[0];
    const float* W     = (const float*)d_in[1];
    const float* b     = (const float*)d_in[2];
    const float* gamma = (const float*)d_in[3];
    const float* beta  = (const float*)d_in[4];
    const float* A     = (const float*)d_in[5];

    float* ws      = (float*)d_ws;
    float* dvec    = ws;                         // NN floats
    float* X       = ws + 16384;                 // NN*64
    float* Yt      = X  + (size_t)NN * DD;       // NN*64 (transposed, scaled)
    float* out_pre = Yt + (size_t)NN * DD;       // NN*64
    float* scale   = out_pre + (size_t)NN * DD;  // 64
    float* shift   = scale + 64;                 // 64

    k_rowsum<<<NN,          256, 0, stream>>>(A, dvec);
    k_linear<<<NN / 4,      256, 0, stream>>>(H, W, b, dvec, X, Yt);
    k_gemm  <<<NN / 32,     256, 0, stream>>>(A, Yt, X, dvec, out_pre);
    k_stats <<<DD,          256, 0, stream>>>(out_pre, gamma, beta, scale, shift);
    k_bn    <<<(NN * DD / 4) / 256, 256, 0, stream>>>(out_pre, scale, shift, (float*)d_out);
}